// HypergraphImageClassifier_7541962571985
// MI455X (gfx1250) — compile-verified
//
#include <hip/hip_runtime.h>
#include <cstdint>
#include <cstddef>

typedef float v2f __attribute__((ext_vector_type(2)));
typedef float v8f __attribute__((ext_vector_type(8)));

namespace hgc {

constexpr int   B    = 4;
constexpr int   NPIX = 4096;     // 64*64
constexpr int   NE   = 600;      // 100+200+300 hyperedges
constexpr int   M    = B * NPIX; // GEMM rows
constexpr float EPSC = 1e-6f;
constexpr float A0   = 0.57735026918962576f;            // 1/sqrt(3) = 1/sqrt(Dv)
constexpr float CV   = EPSC / (1.0f + EPSC * (float)NPIX * A0); // Sherman-Morrison c for Av
constexpr float CVA2 = CV * A0 * A0;

// ---------------------------------------------------------------- utilities
__global__ void zero_int_kernel(int* p, int n) {
    int i = blockIdx.x * blockDim.x + threadIdx.x;
    if (i < n) p[i] = 0;
}

// Count edge degrees De via integer atomics (deterministic result).
__global__ void degree_count_kernel(const int* __restrict__ s100,
                                    const int* __restrict__ s200,
                                    const int* __restrict__ s300,
                                    int* __restrict__ DeInt) {
    int idx = blockIdx.x * blockDim.x + threadIdx.x;
    if (idx >= B * NPIX) return;
    int b = idx / NPIX;
    int n = idx - b * NPIX;
    atomicAdd(&DeInt[b * NE +       s100[b * NPIX + n]], 1);
    atomicAdd(&DeInt[b * NE + 100 + s200[b * NPIX + n]], 1);
    atomicAdd(&DeInt[b * NE + 300 + s300[b * NPIX + n]], 1);
}

// Per-batch: DeF = float(De); ae = 1/De; ce = eps/(1+eps*sum(ae)).
__global__ void edge_params_kernel(const int* __restrict__ DeInt,
                                   float* __restrict__ DeF,
                                   float* __restrict__ ae,
                                   float* __restrict__ ce) {
    int b = blockIdx.x;
    __shared__ float red[256];
    float local = 0.0f;
    for (int e = threadIdx.x; e < NE; e += 256) {
        int   d  = DeInt[b * NE + e];
        float df = (float)d;
        float a  = (d > 0) ? (1.0f / df) : 0.0f;  // setup guarantees d>=1
        DeF[b * NE + e] = df;
        ae [b * NE + e] = a;
        local += a;
    }
    red[threadIdx.x] = local;
    __syncthreads();
    for (int s = 128; s > 0; s >>= 1) {
        if (threadIdx.x < s) red[threadIdx.x] += red[threadIdx.x + s];
        __syncthreads();
    }
    if (threadIdx.x == 0) ce[b] = EPSC / (1.0f + EPSC * red[0]);
}

// ------------------------------------------------- segmented edge sum (H^T)
// edgeSum[b,e,c] = sum over nodes with matching label of x[b,n,c].
// Deterministic: one block per (e,b), sequential scan over nodes.
// Labels are staged through LDS once per block (16KB) so the 4096-iteration
// match scan hits LDS instead of issuing per-n global loads.
template <int C>
__global__ __launch_bounds__(256)
void edge_sum_kernel(const float* __restrict__ x,     // [B,NPIX,C]
                     const int*  __restrict__ s100,
                     const int*  __restrict__ s200,
                     const int*  __restrict__ s300,
                     float* __restrict__ edgeSum) {   // [B,NE,C]
    constexpr int T   = 256;
    constexpr int PER = C / T;
    __shared__ int slab[NPIX];

    int b = blockIdx.y;
    int e = blockIdx.x;
    const int* seg;
    int lbl;
    if (e < 100)      { seg = s100 + b * NPIX; lbl = e; }
    else if (e < 300) { seg = s200 + b * NPIX; lbl = e - 100; }
    else              { seg = s300 + b * NPIX; lbl = e - 300; }

    for (int i = threadIdx.x; i < NPIX; i += T) slab[i] = seg[i];
    __syncthreads();

    float acc[PER];
#pragma unroll
    for (int i = 0; i < PER; ++i) acc[i] = 0.0f;

    const float* xb = x + (size_t)b * NPIX * C;
    for (int n = 0; n < NPIX; ++n) {
        if (slab[n] == lbl) {
            const float* row = xb + (size_t)n * C;
#pragma unroll
            for (int i = 0; i < PER; ++i) acc[i] += row[threadIdx.x + i * T];
        }
    }
    float* out = edgeSum + ((size_t)b * NE + e) * C;
#pragma unroll
    for (int i = 0; i < PER; ++i) out[threadIdx.x + i * T] = acc[i];
}

// ----------------------------------------- edge-space transform (Av,Be fold)
// In:  edgeSum (raw segmented sums of x).  Out (in place): ze2 = Be(H^T(Av x)).
// Also emits sn[b,c] = sum_e De_e * ze2[e,c]  (column sum of H @ ze2).
__global__ void edge_transform_kernel(float* __restrict__ edge,  // [B,NE,C] in/out
                                      const float* __restrict__ DeF,
                                      const float* __restrict__ aeArr,
                                      const float* __restrict__ ceArr,
                                      float* __restrict__ sn,    // [B,C]
                                      int C) {
    int b = blockIdx.y;
    int c = blockIdx.x * blockDim.x + threadIdx.x;
    if (c >= C) return;
    float* Ee = edge + (size_t)b * NE * C;
    const float* De = DeF   + b * NE;
    const float* av = aeArr + b * NE;
    float ce = ceArr[b];

    // colsum over nodes of x == sum of scale-100 edge sums (they partition nodes)
    float colsum = 0.0f;
    for (int e = 0; e < 100; ++e) colsum += Ee[(size_t)e * C + c];
    float k1 = CVA2 * colsum;

    // ze = a0*edgeSum - De*cv*a0^2*colsum ; accumulate se = sum ae*ze
    float se = 0.0f;
    for (int e = 0; e < NE; ++e) {
        float ze = A0 * Ee[(size_t)e * C + c] - De[e] * k1;
        Ee[(size_t)e * C + c] = ze;
        se += av[e] * ze;
    }
    // ze2 = ae*ze - ce*ae*se ; accumulate sn = sum De*ze2
    float snv = 0.0f;
    for (int e = 0; e < NE; ++e) {
        float z = av[e] * (Ee[(size_t)e * C + c] - ce * se);
        Ee[(size_t)e * C + c] = z;
        snv += De[e] * z;
    }
    sn[b * C + c] = snv;
}

// ------------------------------------------------- node gather + final Av
// zfin[b,n,c] = a0*(ze2[e1]+ze2[e2]+ze2[e3]) - cv*a0^2*sn[b,c]
__global__ void gather_nodes_kernel(const float* __restrict__ edge, // ze2 [B,NE,C]
                                    const float* __restrict__ sn,   // [B,C]
                                    const int* __restrict__ s100,
                                    const int* __restrict__ s200,
                                    const int* __restrict__ s300,
                                    float* __restrict__ zfin,       // [B,NPIX,C]
                                    int C) {
    int b = blockIdx.y;
    int n = blockIdx.x;
    int e1 =       s100[b * NPIX + n];
    int e2 = 100 + s200[b * NPIX + n];
    int e3 = 300 + s300[b * NPIX + n];
    const float* Ee = edge + (size_t)b * NE * C;
    const float* r1 = Ee + (size_t)e1 * C;
    const float* r2 = Ee + (size_t)e2 * C;
    const float* r3 = Ee + (size_t)e3 * C;
    const float* sb = sn + b * C;
    float* out = zfin + ((size_t)b * NPIX + n) * C;
    for (int c = threadIdx.x; c < C; c += blockDim.x)
        out[c] = A0 * (r1[c] + r2[c] + r3[c]) - CVA2 * sb[c];
}

// ------------------------------------------------------ fp32 WMMA GEMM
// Y[M,NOUT] = A[M,K] @ Wt[K,NOUT] + bias  (optional ReLU).
// Block = 256 threads = 8 waves; wave handles one 16-col tile x four 16-row
// tiles.  Software-pipelined one stage deep: fragments for step kb+4 are
// loaded before the WMMAs of step kb issue, so global loads overlap matrix
// ops instead of serializing on s_wait_loadcnt 0.
template <int K, int NOUT, bool RELU>
__global__ __launch_bounds__(256)
void wmma_gemm_kernel(const float* __restrict__ A,
                      const float* __restrict__ Wt,
                      const float* __restrict__ bias,
                      float* __restrict__ Y) {
    const int wave = threadIdx.x >> 5;
    const int lane = threadIdx.x & 31;
    const int hi   = lane >> 4;          // half-wave select
    const int lm   = lane & 15;
    const int jb   = (blockIdx.y * 8 + wave) * 16;  // output col base
    const int mb   = blockIdx.x * 64;               // output row base

    // Per-lane base pointers; half-wave K offset (2*hi) folded in once.
    const float* ap0 = A + (size_t)(mb      + lm) * K + 2 * hi;
    const float* ap1 = A + (size_t)(mb + 16 + lm) * K + 2 * hi;
    const float* ap2 = A + (size_t)(mb + 32 + lm) * K + 2 * hi;
    const float* ap3 = A + (size_t)(mb + 48 + lm) * K + 2 * hi;
    const float* bp  = Wt + (size_t)(2 * hi) * NOUT + jb + lm;

    v8f acc0 = {}, acc1 = {}, acc2 = {}, acc3 = {};

    // prologue: load stage kb = 0
    v2f bf, af0, af1, af2, af3;
    bf.x = bp[0];
    bf.y = bp[NOUT];
    af0  = *(const v2f*)(ap0);
    af1  = *(const v2f*)(ap1);
    af2  = *(const v2f*)(ap2);
    af3  = *(const v2f*)(ap3);

#pragma unroll 8
    for (int kb = 0; kb < K; kb += 4) {
        // issue next stage's loads first (dead on final iteration)
        v2f bfn = bf, an0 = af0, an1 = af1, an2 = af2, an3 = af3;
        if (kb + 4 < K) {
            const int kn = kb + 4;
            bfn.x = bp[kn * NOUT];
            bfn.y = bp[(kn + 1) * NOUT];
            an0   = *(const v2f*)(ap0 + kn);
            an1   = *(const v2f*)(ap1 + kn);
            an2   = *(const v2f*)(ap2 + kn);
            an3   = *(const v2f*)(ap3 + kn);
        }
        if ((kb & 31) == 0) {  // compile-time per unrolled body position
            __builtin_prefetch(ap0 + kb + 128, 0, 1);
            __builtin_prefetch(ap1 + kb + 128, 0, 1);
            __builtin_prefetch(ap2 + kb + 128, 0, 1);
            __builtin_prefetch(ap3 + kb + 128, 0, 1);
        }
        acc0 = __builtin_amdgcn_wmma_f32_16x16x4_f32(false, af0, false, bf, (short)0, acc0, false, false);
        acc1 = __builtin_amdgcn_wmma_f32_16x16x4_f32(false, af1, false, bf, (short)0, acc1, false, false);
        acc2 = __builtin_amdgcn_wmma_f32_16x16x4_f32(false, af2, false, bf, (short)0, acc2, false, false);
        acc3 = __builtin_amdgcn_wmma_f32_16x16x4_f32(false, af3, false, bf, (short)0, acc3, false, false);
        bf = bfn; af0 = an0; af1 = an1; af2 = an2; af3 = an3;
    }

    float bv = bias[jb + lm];
#pragma unroll
    for (int r = 0; r < 8; ++r) {
        int m = r + 8 * hi;  // C/D layout: VGPR r -> M=r (lo half), M=r+8 (hi half)
        float v0 = acc0[r] + bv;
        float v1 = acc1[r] + bv;
        float v2 = acc2[r] + bv;
        float v3 = acc3[r] + bv;
        if (RELU) {
            v0 = fmaxf(v0, 0.0f); v1 = fmaxf(v1, 0.0f);
            v2 = fmaxf(v2, 0.0f); v3 = fmaxf(v3, 0.0f);
        }
        Y[(size_t)(mb +      m) * NOUT + jb + lm] = v0;
        Y[(size_t)(mb + 16 + m) * NOUT + jb + lm] = v1;
        Y[(size_t)(mb + 32 + m) * NOUT + jb + lm] = v2;
        Y[(size_t)(mb + 48 + m) * NOUT + jb + lm] = v3;
    }
}

// ------------------------------------------------------ mean over N, then FC
__global__ void mean_part_kernel(const float* __restrict__ y2,  // [B,NPIX,256]
                                 float* __restrict__ part) {    // [B,32,256]
    int b = blockIdx.y;
    int chunk = blockIdx.x;     // 32 chunks of 128 rows
    int c = threadIdx.x;        // 256 channels
    float s = 0.0f;
    const float* base = y2 + ((size_t)b * NPIX + chunk * 128) * 256;
    for (int n = 0; n < 128; ++n) s += base[(size_t)n * 256 + c];
    part[((size_t)b * 32 + chunk) * 256 + c] = s;
}

__global__ void mean_final_kernel(const float* __restrict__ part,
                                  float* __restrict__ meanb) {   // [B,256]
    int b = blockIdx.x;
    int c = threadIdx.x;
    float s = 0.0f;
    for (int k = 0; k < 32; ++k) s += part[((size_t)b * 32 + k) * 256 + c];
    meanb[b * 256 + c] = s * (1.0f / (float)NPIX);
}

__global__ void fc_kernel(const float* __restrict__ meanb,   // [B,256]
                          const float* __restrict__ fcW,     // [256,1000]
                          const float* __restrict__ fcb,     // [1000]
                          float* __restrict__ out) {         // [B,1000]
    int b = blockIdx.x;
    for (int j = threadIdx.x; j < 1000; j += blockDim.x) {
        float s = fcb[j];
        for (int c = 0; c < 256; ++c)
            s += meanb[b * 256 + c] * fcW[c * 1000 + j];
        out[b * 1000 + j] = s;
    }
}

} // namespace hgc

extern "C" void kernel_launch(void* const* d_in, const int* in_sizes, int n_in,
                              void* d_out, int out_size, void* d_ws, size_t ws_size,
                              hipStream_t stream) {
    using namespace hgc;
    (void)in_sizes; (void)n_in; (void)out_size; (void)ws_size;

    // setup_inputs() dict order: seg100, seg200, seg300, features, W1,b1, W2,b2, fcW,fcb
    const int*   s100 = (const int*)  d_in[0];
    const int*   s200 = (const int*)  d_in[1];
    const int*   s300 = (const int*)  d_in[2];
    const float* feat = (const float*)d_in[3];
    const float* W1   = (const float*)d_in[4];
    const float* b1   = (const float*)d_in[5];
    const float* W2   = (const float*)d_in[6];
    const float* b2   = (const float*)d_in[7];
    const float* fcW  = (const float*)d_in[8];
    const float* fcb  = (const float*)d_in[9];
    float*       out  = (float*)d_out;

    // workspace layout (float element offsets)
    float* w     = (float*)d_ws;
    int*   DeInt = (int*)d_ws;                 // B*600 ints
    float* DeF   = w + 2400;                   // B*600
    float* ae    = w + 4800;                   // B*600
    float* ce    = w + 7200;                   // B
    float* sn    = w + 7232;                   // B*1024 (layer2 uses first B*512)
    float* edge  = w + 11328;                  // B*600*1024 (layer2 reuses as B*600*512)
    float* zfin  = w + 2468928;                // B*NPIX*1024 (layer2 reuses as B*NPIX*512)
    float* y1    = w + 19246144;               // B*NPIX*512
    float* y2    = w + 27634752;               // B*NPIX*256
    float* part  = w + 31829056;               // B*32*256
    float* meanb = w + 31861824;               // B*256

    // ---- degrees & edge normalization params
    zero_int_kernel<<<(B * NE + 255) / 256, 256, 0, stream>>>(DeInt, B * NE);
    degree_count_kernel<<<(B * NPIX + 255) / 256, 256, 0, stream>>>(s100, s200, s300, DeInt);
    edge_params_kernel<<<B, 256, 0, stream>>>(DeInt, DeF, ae, ce);

    // ---- layer 1: C = 1024 -> 512, ReLU
    edge_sum_kernel<1024><<<dim3(NE, B), 256, 0, stream>>>(feat, s100, s200, s300, edge);
    edge_transform_kernel<<<dim3(1024 / 256, B), 256, 0, stream>>>(edge, DeF, ae, ce, sn, 1024);
    gather_nodes_kernel<<<dim3(NPIX, B), 256, 0, stream>>>(edge, sn, s100, s200, s300, zfin, 1024);
    wmma_gemm_kernel<1024, 512, true><<<dim3(M / 64, 512 / 128), 256, 0, stream>>>(zfin, W1, b1, y1);

    // ---- layer 2: C = 512 -> 256
    edge_sum_kernel<512><<<dim3(NE, B), 256, 0, stream>>>(y1, s100, s200, s300, edge);
    edge_transform_kernel<<<dim3(512 / 256, B), 256, 0, stream>>>(edge, DeF, ae, ce, sn, 512);
    gather_nodes_kernel<<<dim3(NPIX, B), 256, 0, stream>>>(edge, sn, s100, s200, s300, zfin, 512);
    wmma_gemm_kernel<512, 256, false><<<dim3(M / 64, 256 / 128), 256, 0, stream>>>(zfin, W2, b2, y2);

    // ---- mean over nodes + classifier head
    mean_part_kernel<<<dim3(32, B), 256, 0, stream>>>(y2, part);
    mean_final_kernel<<<B, 256, 0, stream>>>(part, meanb);
    fc_kernel<<<B, 256, 0, stream>>>(meanb, fcW, fcb, out);
}